// GATModel_87402584473800
// MI455X (gfx1250) — compile-verified
//
#include <hip/hip_runtime.h>
#include <math.h>

// ---------------- problem constants (match reference) ----------------
#define NN      50000
#define EE      800000
#define EP      (EE + NN)      // edges incl. self loops
#define HH      4
#define CC      64
#define HC      (HH * CC)      // 256
#define SLOPE   0.2f

typedef float v2f __attribute__((ext_vector_type(2)));
typedef float v8f __attribute__((ext_vector_type(8)));

// ---------------- helpers ----------------
__device__ __forceinline__ void atomicMaxF(float* addr, float v) {
    // works for mixed signs when the location is initialized to -inf
    if (v >= 0.0f) atomicMax((int*)addr, __float_as_int(v));
    else           atomicMin((unsigned int*)addr, __float_as_uint(v));
}

__device__ __forceinline__ void decode_edge(int e, const int* __restrict__ ei,
                                            int& s, int& d) {
    if (e < EE) { s = ei[e]; d = ei[EE + e]; }
    else        { s = e - EE; d = s; }       // self loop
}

// ---------------- per-layer kernels ----------------

// zero agg, set m=-inf, denom=0
__global__ void init_layer(float* __restrict__ agg, float* __restrict__ m,
                           float* __restrict__ denom) {
    int i = blockIdx.x * blockDim.x + threadIdx.x;
    if (i < NN * HC) agg[i] = 0.0f;
    if (i < NN * HH) { m[i] = -INFINITY; denom[i] = 0.0f; }
}

// Hout[N,256] = X[N,fin] @ W[fin,256] via V_WMMA_F32_16X16X4_F32.
// One wave (32 threads) per 16x16 output tile. N=50000 is divisible by 16.
// A layout (16x4 f32): lane r (0-15) holds row M=r, {K=0,K=1} in {v0,v1};
//                      lane r+16 holds row M=r, {K=2,K=3}.
// B layout (4x16 f32): lanes 0-15 hold N=lane with {K=0,K=1}; lanes 16-31 {K=2,K=3}.
// C/D layout: VGPR j -> M = j (+8 for upper lane half), N = lane&15.
__global__ void gemm_wmma_f32(const float* __restrict__ X,
                              const float* __restrict__ W,
                              float* __restrict__ Hout, int fin) {
    const int lane = threadIdx.x;
    const int half = lane >> 4;            // 0 / 1
    const int r    = lane & 15;
    const int m0   = blockIdx.x * 16;      // M tile
    const int n0   = blockIdx.y * 16;      // N tile of 256
    const float* xrow = X + (size_t)(m0 + r) * fin;
    v8f acc = {0.f,0.f,0.f,0.f,0.f,0.f,0.f,0.f};
    for (int k0 = 0; k0 < fin; k0 += 4) {
        const int ka = k0 + 2 * half;
        v2f a = *(const v2f*)(xrow + ka);                 // b64 load, K,K+1
        v2f b;
        b.x = W[(size_t)ka * HC + n0 + r];
        b.y = W[(size_t)(ka + 1) * HC + n0 + r];
        acc = __builtin_amdgcn_wmma_f32_16x16x4_f32(
                  false, a, false, b, (short)0, acc, false, false);
    }
    float* out = Hout + (size_t)(m0 + half * 8) * HC + n0 + r;
#pragma unroll
    for (int j = 0; j < 8; ++j) out[(size_t)j * HC] = acc[j];
}

// alpha_src[n,h] = dot(h[n,h,:], a_s[h,:]); alpha_dst likewise
__global__ void alpha_kernel(const float* __restrict__ hfeat,
                             const float* __restrict__ a_s,
                             const float* __restrict__ a_d,
                             float* __restrict__ asrc,
                             float* __restrict__ adst) {
    int i = blockIdx.x * blockDim.x + threadIdx.x;
    if (i >= NN * HH) return;
    int n = i >> 2, h = i & 3;
    const float* hp = hfeat + (size_t)n * HC + h * CC;
    const float* ws = a_s + h * CC;
    const float* wd = a_d + h * CC;
    float s1 = 0.f, s2 = 0.f;
#pragma unroll 4
    for (int c = 0; c < CC; ++c) { float v = hp[c]; s1 += v * ws[c]; s2 += v * wd[c]; }
    asrc[i] = s1;
    adst[i] = s2;
}

__device__ __forceinline__ float edge_logit(const float* asrc, const float* adst,
                                            int s, int d, int h) {
    float v = asrc[s * HH + h] + adst[d * HH + h];
    return v > 0.0f ? v : SLOPE * v;     // leaky_relu
}

// pass 1: scatter-max of logits into m[dst,h]
__global__ void edge_max(const int* __restrict__ ei,
                         const float* __restrict__ asrc,
                         const float* __restrict__ adst,
                         float* __restrict__ m) {
    int i = blockIdx.x * blockDim.x + threadIdx.x;
    if (i >= EP * HH) return;
    int e = i >> 2, h = i & 3, s, d;
    decode_edge(e, ei, s, d);
    atomicMaxF(&m[d * HH + h], edge_logit(asrc, adst, s, d, h));
}

// pass 2: denom[dst,h] += exp(e - m[dst,h])
__global__ void edge_denom(const int* __restrict__ ei,
                           const float* __restrict__ asrc,
                           const float* __restrict__ adst,
                           const float* __restrict__ m,
                           float* __restrict__ denom) {
    int i = blockIdx.x * blockDim.x + threadIdx.x;
    if (i >= EP * HH) return;
    int e = i >> 2, h = i & 3, s, d;
    decode_edge(e, ei, s, d);
    float ex = __expf(edge_logit(asrc, adst, s, d, h) - m[d * HH + h]);
    atomicAdd(&denom[d * HH + h], ex);
}

// pass 3: agg[dst,h,:] += alpha * h[src,h,:]   (wave per edge)
__global__ void edge_agg(const int* __restrict__ ei,
                         const float* __restrict__ asrc,
                         const float* __restrict__ adst,
                         const float* __restrict__ m,
                         const float* __restrict__ denom,
                         const float* __restrict__ hfeat,
                         float* __restrict__ agg) {
    int wid  = (blockIdx.x * blockDim.x + threadIdx.x) >> 5;   // global wave = edge
    int lane = threadIdx.x & 31;
    if (wid >= EP) return;
    int s, d;
    decode_edge(wid, ei, s, d);
    float al[HH];
#pragma unroll
    for (int h = 0; h < HH; ++h) {
        float ex = __expf(edge_logit(asrc, adst, s, d, h) - m[d * HH + h]);
        al[h] = ex / (denom[d * HH + h] + 1e-16f);
    }
    const float* hs = hfeat + (size_t)s * HC;
    float* ag = agg + (size_t)d * HC;
#pragma unroll
    for (int h = 0; h < HH; ++h) {
        float a = al[h];
        atomicAdd(&ag[h * CC + lane],      a * hs[h * CC + lane]);
        atomicAdd(&ag[h * CC + 32 + lane], a * hs[h * CC + 32 + lane]);
    }
}

// head mean + bias + relu -> layer output [N,64]
__global__ void mean_relu(const float* __restrict__ agg,
                          const float* __restrict__ b,
                          float* __restrict__ out) {
    int i = blockIdx.x * blockDim.x + threadIdx.x;
    if (i >= NN * CC) return;
    int n = i >> 6, c = i & 63;
    const float* ag = agg + (size_t)n * HC;
    float v = 0.25f * (ag[c] + ag[CC + c] + ag[2 * CC + c] + ag[3 * CC + c]) + b[c];
    out[i] = v > 0.0f ? v : 0.0f;
}

// ---------------- host driver ----------------
extern "C" void kernel_launch(void* const* d_in, const int* in_sizes, int n_in,
                              void* d_out, int out_size, void* d_ws, size_t ws_size,
                              hipStream_t stream) {
    const float* x  = (const float*)d_in[0];
    const int*   ei = (const int*)d_in[1];
    const float* Wm[3]   = {(const float*)d_in[2], (const float*)d_in[6],  (const float*)d_in[10]};
    const float* As[3]   = {(const float*)d_in[3], (const float*)d_in[7],  (const float*)d_in[11]};
    const float* Ad[3]   = {(const float*)d_in[4], (const float*)d_in[8],  (const float*)d_in[12]};
    const float* Bias[3] = {(const float*)d_in[5], (const float*)d_in[9],  (const float*)d_in[13]};

    // workspace layout (floats)
    float* ws     = (float*)d_ws;
    float* hfeat  = ws;                       // N*256
    float* agg    = hfeat + (size_t)NN * HC;  // N*256
    float* asrc   = agg   + (size_t)NN * HC;  // N*H
    float* adst   = asrc  + (size_t)NN * HH;  // N*H
    float* mbuf   = adst  + (size_t)NN * HH;  // N*H
    float* denom  = mbuf  + (size_t)NN * HH;  // N*H
    float* xbufA  = denom + (size_t)NN * HH;  // N*64
    float* xbufB  = xbufA + (size_t)NN * CC;  // N*64

    const float* xin = x;
    int fin = 128;
    float* lout[3] = {xbufA, xbufB, (float*)d_out};

    const int TB = 256;
    dim3 gInit((NN * HC + TB - 1) / TB);
    dim3 gGemm(NN / 16, HC / 16);
    dim3 gAlpha((NN * HH + TB - 1) / TB);
    dim3 gEdge((EP * HH + TB - 1) / TB);
    dim3 gAgg(((size_t)EP * 32 + TB - 1) / TB);
    dim3 gMean((NN * CC + TB - 1) / TB);

    for (int L = 0; L < 3; ++L) {
        init_layer   <<<gInit,  TB, 0, stream>>>(agg, mbuf, denom);
        gemm_wmma_f32<<<gGemm,  32, 0, stream>>>(xin, Wm[L], hfeat, fin);
        alpha_kernel <<<gAlpha, TB, 0, stream>>>(hfeat, As[L], Ad[L], asrc, adst);
        edge_max     <<<gEdge,  TB, 0, stream>>>(ei, asrc, adst, mbuf);
        edge_denom   <<<gEdge,  TB, 0, stream>>>(ei, asrc, adst, mbuf, denom);
        edge_agg     <<<gAgg,   TB, 0, stream>>>(ei, asrc, adst, mbuf, denom, hfeat, agg);
        mean_relu    <<<gMean,  TB, 0, stream>>>(agg, Bias[L], lout[L]);
        xin = lout[L];
        fin = CC;
    }
}